// PathFinder_9964324127492
// MI455X (gfx1250) — compile-verified
//
#include <hip/hip_runtime.h>
#include <stdint.h>

// Problem geometry (fixed by the reference).
#define LVLS   64
#define PNODES 32768
#define EPER   262144

// Scatter tiling: 256 blocks/level, 1024 edges/block, 4 edges/thread.
#define TILE        1024
#define SCATTER_BLK 256

typedef unsigned int v4u __attribute__((ext_vector_type(4)));
typedef int          v4i __attribute__((ext_vector_type(4)));
typedef int          v8i __attribute__((ext_vector_type(8)));

#if defined(__AMDGCN__) && __has_builtin(__builtin_amdgcn_tensor_load_to_lds)
#define USE_TDM 1
#else
#define USE_TDM 0
#endif

#if USE_TDM
// Issue one TDM descriptor: 1-D tensor of `elems` 4-byte words at `gptr`,
// tile of `tile_elems` words, landing at LDS byte address `lds_addr`.
// D# layout per CDNA5 ISA ch.8 (group0: count/lds/global/type; group1:
// data_size, tensor_dim0, tile_dim0, dim0 stride). Groups 2/3 zero (<=2D).
__device__ __forceinline__ void tdm_load_1d(uint32_t lds_addr, const void* gptr,
                                            uint32_t elems, uint32_t tile_elems) {
  uint64_t ga = (uint64_t)(uintptr_t)gptr;
  v4u g0;
  g0[0] = 1u;                                                   // count=1 (valid), user mode
  g0[1] = lds_addr;                                             // lds_addr[31:0]
  g0[2] = (uint32_t)ga;                                         // global_addr[31:0]
  g0[3] = ((uint32_t)(ga >> 32) & 0x01FFFFFFu) | (2u << 30);    // global_addr[56:32] | type=2
  v8i g1;
  g1[0] = 0x00020000;                                           // data_size=2 (4 bytes)
  g1[1] = (int)((elems & 0xFFFFu) << 16);                       // tensor_dim0[15:0]
  g1[2] = (int)(((elems >> 16) & 0xFFFFu) | (1u << 16));        // tensor_dim0[31:16], tensor_dim1=1
  g1[3] = (int)((tile_elems & 0xFFFFu) << 16);                  // tile_dim0
  g1[4] = 0;                                                    // tile_dim1/2 unused
  g1[5] = (int)elems;                                           // tensor_dim0_stride[31:0]
  g1[6] = 0;
  g1[7] = 0;
  v4i z4 = {0, 0, 0, 0};
#if __clang_major__ >= 23
  v8i z8 = {0, 0, 0, 0, 0, 0, 0, 0};
  __builtin_amdgcn_tensor_load_to_lds(g0, g1, z4, z4, z8, 0);
#else
  __builtin_amdgcn_tensor_load_to_lds(g0, g1, z4, z4, 0);
#endif
}
#endif

// Zero the candidate buffer (sentinel 0u == 0.0f; all real candidates >= 1.0f).
__global__ __launch_bounds__(256) void init_cand(unsigned* __restrict__ cand) {
  int i = blockIdx.x * blockDim.x + threadIdx.x;
  cand[i] = 0u;
}

// One level's edge scatter: stage the (src,dst) tile into LDS via the Tensor
// Data Mover, then gather depths and scatter-max into cand with u32 atomics
// (bit-pattern order == IEEE order for the non-negative depths here).
__global__ __launch_bounds__(SCATTER_BLK)
void scatter_level(const int* __restrict__ src, const int* __restrict__ dst,
                   const float* __restrict__ h, unsigned* __restrict__ cand,
                   int lvl_base) {
  __shared__ int smem[2 * TILE];            // [0,TILE): src tile, [TILE,2T): dst tile
  const int tile0 = blockIdx.x * TILE;
  const uint32_t lds_base = (uint32_t)(uintptr_t)&smem[0];  // LDS byte offset (low 32 bits)

#if USE_TDM
  if ((threadIdx.x >> 5) == 0) {            // wave 0 issues the DMA (TDM ignores EXEC)
    tdm_load_1d(lds_base,                   (const void*)(src + tile0), TILE, TILE);
    tdm_load_1d(lds_base + TILE * 4u,       (const void*)(dst + tile0), TILE, TILE);
    __builtin_amdgcn_s_wait_tensorcnt(0);
  }
  // The TDM builtin takes the LDS address as *data*, so the compiler cannot
  // see that smem is written. This asm is a conservative "smem may have been
  // written through this address" barrier: without it, the ds_loads below are
  // reads of a never-stored __shared__ array and get folded to undef.
  asm volatile("" : : "s"(lds_base) : "memory");
  __syncthreads();
#else
  for (int i = threadIdx.x; i < TILE; i += (int)blockDim.x) {
    smem[i]        = __builtin_nontemporal_load(src + tile0 + i);
    smem[TILE + i] = __builtin_nontemporal_load(dst + tile0 + i);
  }
  __syncthreads();
#endif

#pragma unroll
  for (int k = 0; k < TILE / SCATTER_BLK; ++k) {
    const int i = threadIdx.x + k * SCATTER_BLK;
    int s = smem[i];
    int d = smem[TILE + i];
    float v = h[s] + 1.0f;                  // depth candidate, always >= 1.0
    atomicMax(&cand[d - lvl_base], __float_as_uint(v));
  }
}

// Select: nodes with an in-edge (cand bits != 0) take the new depth, others
// keep the copied-through hdr value. Also resets cand for the next level.
__global__ __launch_bounds__(256)
void merge_reset(float* __restrict__ out_lvl, unsigned* __restrict__ cand) {
  int i = blockIdx.x * blockDim.x + threadIdx.x;
  unsigned c = cand[i];
  if (c != 0u) out_lvl[i] = __uint_as_float(c);
  cand[i] = 0u;
}

extern "C" void kernel_launch(void* const* d_in, const int* in_sizes, int n_in,
                              void* d_out, int out_size, void* d_ws, size_t ws_size,
                              hipStream_t stream) {
  const float* hdr = (const float*)d_in[0];
  const int*   src = (const int*)d_in[1];   // [63, EPER]
  const int*   dst = (const int*)d_in[2];   // [63, EPER]
  float*       out  = (float*)d_out;        // N = LVLS*PNODES depths, updated in place
  unsigned*    cand = (unsigned*)d_ws;      // PNODES words of scratch (128 KB)

  // Pass-through copy: levels/nodes with no in-edges keep their input depth.
  hipMemcpyAsync(out, hdr, (size_t)LVLS * PNODES * sizeof(float),
                 hipMemcpyDeviceToDevice, stream);
  hipLaunchKernelGGL(init_cand, dim3(PNODES / 256), dim3(256), 0, stream, cand);

  // 63 serially dependent wavefront steps; stream order provides the
  // scatter -> merge -> next-level-scatter dependency chain.
  for (int l = 1; l < LVLS; ++l) {
    hipLaunchKernelGGL(scatter_level, dim3(EPER / TILE), dim3(SCATTER_BLK), 0, stream,
                       src + (size_t)(l - 1) * EPER,
                       dst + (size_t)(l - 1) * EPER,
                       out, cand, l * PNODES);
    hipLaunchKernelGGL(merge_reset, dim3(PNODES / 256), dim3(256), 0, stream,
                       out + (size_t)l * PNODES, cand);
  }
}